// Multihead_RGATLayer_62483184222293
// MI455X (gfx1250) — compile-verified
//
#include <hip/hip_runtime.h>
#include <hip/hip_bf16.h>
#include <math.h>

// Problem constants (match reference)
#define NN 50000
#define EE 800000
#define RR 8
#define IND 128
#define OUTD 128
#define HH 8
#define HDD 16

typedef float v2f __attribute__((ext_vector_type(2)));
typedef float v8f __attribute__((ext_vector_type(8)));

// ---------------------------------------------------------------------------
// Sign-split atomic float max. Valid with buffer initialized to -inf:
//  - non-negative values: int-compare order == float order
//  - negative values: unsigned order is reversed float order -> use umin
// ---------------------------------------------------------------------------
__device__ __forceinline__ void atomicMaxFloat(float* addr, float v) {
    if (v >= 0.0f) {
        atomicMax((int*)addr, __float_as_int(v));
    } else {
        atomicMin((unsigned int*)addr, __float_as_uint(v));
    }
}

// ---------------------------------------------------------------------------
// Init: zero d_out (accumulated num) and denom, set running max to -inf.
// ---------------------------------------------------------------------------
__global__ __launch_bounds__(256) void rgat_init(float* __restrict__ out,
                                                 float* __restrict__ maxbuf,
                                                 float* __restrict__ denom) {
    int t = blockIdx.x * 256 + threadIdx.x;
    if (t < NN * OUTD) out[t] = 0.0f;
    if (t < NN * HH) {
        denom[t] = 0.0f;
        maxbuf[t] = -INFINITY;
    }
}

// ---------------------------------------------------------------------------
// Kernel A: z[r, n, :] = h[n, :] @ W[r]  via V_WMMA_F32_16X16X4_F32.
// Block = 256 threads (8 waves), handles one relation r and 64 node rows.
//
// LDS:
//   h_s : 64 x 128 h tile, row stride HS_A=132 (A reads conflict-free: bank
//         step 4 per lane over 16 lanes).
//   w_s : W_r in K-pair-interleaved, fragment-ready layout:
//            w_s[kp*WPS + col*2 + j] = W[2*kp + j][col]   (kp = 0..63, j = 0..1)
//         so a B fragment (K = k+2g, k+2g+1 at column col) is ONE contiguous
//         ds_load_b64 at (k/2 + g)*WPS + col*2 — no register marshaling.
//         WPS=288 -> WPS%64==32: g=1 half-wave uses the bank window disjoint
//         from g=0 -> conflict-free b64 reads (32 lanes x 2 banks = 64 banks).
//
// Fragment layouts per CDNA5 ISA 7.12.2 (32-bit A 16x4 / B 4x16 / C 16x16):
//   lane = (ln = lane&15, g = lane>>4)
//   A vgpr v : element (M=ln,       K = k + 2g + v)
//   B vgpr v : element (K = k+2g+v, N = ln)
//   C vgpr v : element (M = 8g + v, N = ln)
// ---------------------------------------------------------------------------
#define HS_A 132
#define WPS  288   // floats per K-pair row of w_s (256 data + 32 pad)

__global__ __launch_bounds__(256) void rgat_proj_wmma(const float* __restrict__ h,
                                                      const float* __restrict__ W,
                                                      float* __restrict__ z) {
    extern __shared__ float lds[];
    float* h_s = lds;                 // [64][HS_A]
    float* w_s = lds + 64 * HS_A;     // [64][WPS]

    const int r     = blockIdx.y;
    const int nbase = blockIdx.x * 64;
    const int tid   = threadIdx.x;

    // Stage h tile: 64 rows x 128 cols (clamp OOB rows; results masked at store)
    for (int i = tid; i < 64 * 32; i += 256) {          // i indexes float4 slots
        int row = i >> 5;
        int c4  = i & 31;
        int g   = nbase + row;
        if (g >= NN) g = NN - 1;
        float4 v = *(const float4*)(h + (size_t)g * IND + c4 * 4);
        float* d = h_s + row * HS_A + c4 * 4;
        d[0] = v.x; d[1] = v.y; d[2] = v.z; d[3] = v.w;
    }
    // Stage W_r in K-pair-interleaved layout: 64 kp-rows x 128 cols x 2.
    // Consecutive threads take consecutive cols -> coalesced global reads,
    // conflict-free b64 LDS stores.
    const float* Wr = W + (size_t)r * IND * OUTD;
    for (int i = tid; i < 64 * 128; i += 256) {
        int kp  = i >> 7;         // K-pair index 0..63
        int col = i & 127;
        float w0 = Wr[(size_t)(2 * kp) * OUTD + col];
        float w1 = Wr[(size_t)(2 * kp + 1) * OUTD + col];
        v2f p; p.x = w0; p.y = w1;
        *(v2f*)(w_s + kp * WPS + col * 2) = p;
    }
    __syncthreads();

    const int wave = tid >> 5;
    const int lane = tid & 31;
    const int ln   = lane & 15;
    const int g    = lane >> 4;
    const int rt   = wave & 3;          // 16-row tile within the 64-row block
    const int cb   = (wave >> 2) * 64;  // column base (two 64-wide halves)

    v8f acc0 = {}, acc1 = {}, acc2 = {}, acc3 = {};
    const float* arow = h_s + (rt * 16 + ln) * HS_A;
    const float* bcol = w_s + (size_t)g * WPS + (size_t)(cb + ln) * 2;

    for (int k = 0; k < IND; k += 4) {
        // A fragment: (M=ln, K=k+2g .. k+2g+1), one 8B LDS read
        v2f a = *(const v2f*)(arow + k + 2 * g);
        const float* brow = bcol + (k >> 1) * WPS;
        // Four B fragments, each one contiguous 8B LDS read (128B apart ->
        // pairable into ds_load_2addr_b64 by the compiler).
        v2f b0 = *(const v2f*)(brow + 0 * 32);
        v2f b1 = *(const v2f*)(brow + 1 * 32);
        v2f b2 = *(const v2f*)(brow + 2 * 32);
        v2f b3 = *(const v2f*)(brow + 3 * 32);
        acc0 = __builtin_amdgcn_wmma_f32_16x16x4_f32(false, a, false, b0, (short)0, acc0, false, false);
        acc1 = __builtin_amdgcn_wmma_f32_16x16x4_f32(false, a, false, b1, (short)0, acc1, false, false);
        acc2 = __builtin_amdgcn_wmma_f32_16x16x4_f32(false, a, false, b2, (short)0, acc2, false, false);
        acc3 = __builtin_amdgcn_wmma_f32_16x16x4_f32(false, a, false, b3, (short)0, acc3, false, false);
    }

    // Store: C vgpr v -> row nbase + rt*16 + 8g + v, col cb + ct*16 + ln
    float* zr = z + (size_t)r * NN * OUTD;
#pragma unroll
    for (int ct = 0; ct < 4; ++ct) {
        v8f acc = (ct == 0) ? acc0 : (ct == 1) ? acc1 : (ct == 2) ? acc2 : acc3;
        int col = cb + ct * 16 + ln;
#pragma unroll
        for (int v = 0; v < 8; ++v) {
            int row = nbase + rt * 16 + 8 * g + v;
            if (row < NN) zr[(size_t)row * OUTD + col] = acc[v];
        }
    }
}

// ---------------------------------------------------------------------------
// Kernel B: per (edge, head) attention logit + leaky_relu + atomic running max.
// ---------------------------------------------------------------------------
__global__ __launch_bounds__(256) void rgat_edge_score(const float* __restrict__ z,
                                                       const float* __restrict__ attw,
                                                       const int* __restrict__ src,
                                                       const int* __restrict__ dst,
                                                       const int* __restrict__ etype,
                                                       float* __restrict__ e_val,
                                                       float* __restrict__ maxbuf) {
    int t = blockIdx.x * 256 + threadIdx.x;
    if (t >= EE * HH) return;
    int e  = t >> 3;
    int hd = t & 7;
    int r = etype[e], s = src[e], d = dst[e];

    const float4* zs = (const float4*)(z + ((size_t)r * NN + s) * OUTD + hd * HDD);
    const float4* zd = (const float4*)(z + ((size_t)r * NN + d) * OUTD + hd * HDD);
    const float4* aw = (const float4*)(attw + ((size_t)r * HH + hd) * (2 * HDD));

    float acc = 0.0f;
#pragma unroll
    for (int q = 0; q < 4; ++q) {
        float4 a = zs[q], wA = aw[q];
        float4 b = zd[q], wB = aw[4 + q];
        acc += a.x * wA.x + a.y * wA.y + a.z * wA.z + a.w * wA.w;
        acc += b.x * wB.x + b.y * wB.y + b.z * wB.z + b.w * wB.w;
    }
    float ev = (acc > 0.0f) ? acc : 0.01f * acc;   // leaky_relu, slope 0.01
    e_val[t] = ev;
    atomicMaxFloat(maxbuf + (size_t)d * HH + hd, ev);
}

// ---------------------------------------------------------------------------
// Kernel C: exp(e - m[dst]), atomic denom, atomic weighted scatter of z_s
// directly into d_out (which accumulates the numerator).
// ---------------------------------------------------------------------------
__global__ __launch_bounds__(256) void rgat_edge_accum(const float* __restrict__ z,
                                                       const int* __restrict__ src,
                                                       const int* __restrict__ dst,
                                                       const int* __restrict__ etype,
                                                       const float* __restrict__ e_val,
                                                       const float* __restrict__ maxbuf,
                                                       float* __restrict__ denom,
                                                       float* __restrict__ out) {
    int t = blockIdx.x * 256 + threadIdx.x;
    if (t >= EE * HH) return;
    int e  = t >> 3;
    int hd = t & 7;
    int r = etype[e], s = src[e], d = dst[e];

    float m  = maxbuf[(size_t)d * HH + hd];       // finite: this edge contributed
    float ex = __expf(e_val[t] - m);
    atomicAdd(denom + (size_t)d * HH + hd, ex);

    const float* zs = z + ((size_t)r * NN + s) * OUTD + hd * HDD;
    float* o = out + (size_t)d * OUTD + hd * HDD;
#pragma unroll
    for (int i = 0; i < HDD; ++i) atomicAdd(o + i, ex * zs[i]);
}

// ---------------------------------------------------------------------------
// Kernel D: out /= max(denom, 1e-9)   (empty nodes -> 0, matches reference)
// ---------------------------------------------------------------------------
__global__ __launch_bounds__(256) void rgat_finalize(float* __restrict__ out,
                                                     const float* __restrict__ denom) {
    int t = blockIdx.x * 256 + threadIdx.x;
    if (t >= NN * OUTD) return;
    out[t] = out[t] / fmaxf(denom[t >> 4], 1e-9f);
}

// ---------------------------------------------------------------------------
extern "C" void kernel_launch(void* const* d_in, const int* in_sizes, int n_in,
                              void* d_out, int out_size, void* d_ws, size_t ws_size,
                              hipStream_t stream) {
    const float* h    = (const float*)d_in[0];   // [N, IN]
    const float* fcw  = (const float*)d_in[1];   // [R, IN, OUT]
    const float* attw = (const float*)d_in[2];   // [R, H, 2*HD]
    const int*   src  = (const int*)d_in[3];     // [E]
    const int*   dst  = (const int*)d_in[4];     // [E]
    const int*   ety  = (const int*)d_in[5];     // [E]
    float* out = (float*)d_out;                  // [N, H, HD]

    // Workspace layout: z | e_val | maxbuf | denom  (~233 MB)
    float* z      = (float*)d_ws;                                   // R*N*OUT
    float* e_val  = z + (size_t)RR * NN * OUTD;                     // E*H
    float* maxbuf = e_val + (size_t)EE * HH;                        // N*H
    float* denom  = maxbuf + (size_t)NN * HH;                       // N*H

    // 1) init accumulators
    {
        int total = NN * OUTD;
        rgat_init<<<(total + 255) / 256, 256, 0, stream>>>(out, maxbuf, denom);
    }
    // 2) per-relation projection GEMM (WMMA f32 16x16x4)
    {
        dim3 grid((NN + 63) / 64, RR);
        size_t shmem = (size_t)(64 * HS_A + 64 * WPS) * sizeof(float); // ~105 KB
        rgat_proj_wmma<<<grid, 256, shmem, stream>>>(h, fcw, z);
    }
    // 3) edge logits + running max
    {
        int total = EE * HH;
        rgat_edge_score<<<(total + 255) / 256, 256, 0, stream>>>(
            z, attw, src, dst, ety, e_val, maxbuf);
    }
    // 4) exp + denom + numerator scatter
    {
        int total = EE * HH;
        rgat_edge_accum<<<(total + 255) / 256, 256, 0, stream>>>(
            z, src, dst, ety, e_val, maxbuf, denom, out);
    }
    // 5) normalize
    {
        int total = NN * OUTD;
        rgat_finalize<<<(total + 255) / 256, 256, 0, stream>>>(out, denom);
    }
}